// VectorQuantizer1D_43885975831076
// MI455X (gfx1250) — compile-verified
//
#include <hip/hip_runtime.h>

// Problem sizes (fixed by the reference)
#define BS_ 32
#define D_  64
#define S_  2048
#define K_  1024
#define N_  (BS_ * S_)   // 65536 flat vectors

typedef __attribute__((ext_vector_type(2))) float v2f;
typedef __attribute__((ext_vector_type(8))) float v8f;
typedef int v4i __attribute__((vector_size(16)));   // matches builtin param type

// ---- output layout (elements, concatenated in reference return order) ----
#define OFF_Q    0
#define SZ_Q     (BS_ * D_ * S_)         // 4194304  quantized_st [B,D,S] f32
#define OFF_LOSS (OFF_Q + SZ_Q)          //          loss scalar f32
#define OFF_OH   (OFF_LOSS + 1)          //          one_hot [B,K,S] i32
#define SZ_OH    (BS_ * K_ * S_)         // 67108864
#define OFF_IDX  (OFF_OH + SZ_OH)        //          encoding_indices [B,S] i32
#define SZ_IDX   (BS_ * S_)
#define OFF_DIST (OFF_IDX + SZ_IDX)      //          distance [B,K,S] f32
#define SZ_DIST  (BS_ * K_ * S_)
#define OFF_UNQ  (OFF_DIST + SZ_DIST)    //          unquantized [N,D] f32
#define SZ_UNQ   (N_ * D_)
#define OFF_EMBT (OFF_UNQ + SZ_UNQ)      //          embeddings.T [K,D] f32

// ---- gfx1250 async global->LDS path (guarded; sync fallback otherwise) ----
#if defined(__AMDGCN__) && __has_builtin(__builtin_amdgcn_global_load_async_to_lds_b128)
#define USE_ASYNC_LDS 1
#else
#define USE_ASYNC_LDS 0
#endif

#if USE_ASYNC_LDS
__device__ __forceinline__ void async_b128_to_lds(const void* g, void* l) {
  // Global generic address == AS1 address numerically; generic LDS address
  // truncated to 32 bits == AS3 byte offset (ISA 10.2 aperture mapping).
  __builtin_amdgcn_global_load_async_to_lds_b128(
      (__attribute__((address_space(1))) v4i*)(unsigned long long)(uintptr_t)g,
      (__attribute__((address_space(3))) v4i*)(unsigned)(uintptr_t)l,
      /*offset=*/0, /*cpol=*/0);
}
__device__ __forceinline__ void wait_async0() {
#if __has_builtin(__builtin_amdgcn_s_wait_asynccnt)
  __builtin_amdgcn_s_wait_asynccnt(0);
#else
  asm volatile("s_wait_asynccnt 0" ::: "memory");
#endif
}
#endif

// ---------------------------------------------------------------------------
// Kernel 1: b2[k] = sum_d emb[d][k]^2 ; write embeddings.T ; zero loss slot
// ---------------------------------------------------------------------------
__global__ void __launch_bounds__(256)
vq_prep(const float* __restrict__ emb, float* __restrict__ out,
        float* __restrict__ b2ws) {
  const int k = blockIdx.x * blockDim.x + threadIdx.x;
  if (k == 0) out[OFF_LOSS] = 0.0f;
  if (k < K_) {
    float* __restrict__ embT = out + OFF_EMBT;
    float s = 0.0f;
#pragma unroll 8
    for (int d = 0; d < D_; ++d) {
      const float e = emb[(size_t)d * K_ + k];
      s += e * e;
      embT[(size_t)k * D_ + d] = e;
    }
    b2ws[k] = s;
  }
}

// ---------------------------------------------------------------------------
// Kernel 2: zero the one_hot region (harness poisons buffers with 0xAA)
// ---------------------------------------------------------------------------
__global__ void __launch_bounds__(256)
vq_zero_onehot(int* __restrict__ oh) {
  const long long i = (long long)blockIdx.x * blockDim.x + threadIdx.x;
  const long long stride = (long long)gridDim.x * blockDim.x;
  for (long long p = i; p < (long long)SZ_OH; p += stride) oh[p] = 0;
}

// ---------------------------------------------------------------------------
// Kernel 3: WMMA distance GEMM + argmin + one_hot/index scatter
//   - 8 waves/block; wave w owns rows n0..n0+15 of flat (= x transposed)
//   - A tile (16x64 f32) lives in 32 VGPRs per lane for the whole kernel
//   - per 16-col panel: double-buffered async global->LDS copy of the
//     64x16 B panel, 16x v_wmma_f32_16x16x4_f32, dist = a2 - 2*dot + b2
// ---------------------------------------------------------------------------
__global__ void __launch_bounds__(256)
vq_dist(const float* __restrict__ x, const float* __restrict__ emb,
        const float* __restrict__ b2ws, float* __restrict__ out) {
  __shared__ float sB[2 * D_ * 16];  // 2 x 4KB B panels: sB[buf*1024 + d*16+col]

  const int lane = threadIdx.x & 31;
  const int wave = threadIdx.x >> 5;
  const int half = lane >> 4;    // 0: K rows {4k,4k+1}, 1: {4k+2,4k+3}
  const int lcol = lane & 15;

  const int tile = blockIdx.x * 8 + wave;   // 0..4095 row tiles
  const int n0   = tile * 16;
  const int b    = n0 / S_;                 // whole tile shares batch index
  const int s0   = n0 % S_;

  // this thread's slice of every B panel: 4 floats (one b128)
  const int pe   = threadIdx.x * 4;         // 256 threads * 4 = 1024 floats
  const int pd   = pe >> 4;                 // d row in panel
  const int pcol = pe & 15;                 // col in panel (multiple of 4)
  const float* __restrict__ gpanel = emb + (size_t)pd * K_ + pcol;

  // ---- load A: flat[n0+row][d], row = lcol; per-lane d = 4*kk + 2*half {+1}
  // x layout [B,D,S] => flat[n][d] = x[b*D*S + d*S + s]
  const float* __restrict__ xb = x + (size_t)b * D_ * S_ + s0 + lcol;
  float a0[16], a1[16];
  float part = 0.0f;
#pragma unroll
  for (int kk = 0; kk < 16; ++kk) {
    const int d0 = 4 * kk + 2 * half;
    const float va = xb[(size_t)d0 * S_];
    const float vb = xb[(size_t)(d0 + 1) * S_];
    a0[kk] = va; a1[kk] = vb;
    part = fmaf(va, va, fmaf(vb, vb, part));
  }
  // a2 of row lcol: combine the two lane halves (d%4<2 and d%4>=2)
  const float a2full = part + __shfl_xor(part, 16, 32);
  // broadcast a2 of the row each C/D register slot maps to: M = r + 8*half
  float a2row[8];
#pragma unroll
  for (int r = 0; r < 8; ++r)
    a2row[r] = __shfl(a2full, r + 8 * half, 32);

  float bestV[8];
  int   bestI[8];
#pragma unroll
  for (int r = 0; r < 8; ++r) { bestV[r] = 3.4e38f; bestI[r] = 0; }

  float* __restrict__ dist = out + OFF_DIST + (size_t)b * K_ * S_;

  // ---- prologue: fetch panel 0 into buffer 0
#if USE_ASYNC_LDS
  async_b128_to_lds(gpanel, sB + pd * 16 + pcol);
  wait_async0();
#else
  *(float4*)(sB + pd * 16 + pcol) = *(const float4*)gpanel;
#endif
  __syncthreads();

  for (int c = 0; c < K_ / 16; ++c) {
    const int c0  = c * 16;
    const int cur = c & 1;

    // prefetch next panel into the other buffer (overlaps WMMA + stores)
    if (c + 1 < K_ / 16) {
#if USE_ASYNC_LDS
      async_b128_to_lds(gpanel + (c + 1) * 16,
                        sB + (cur ^ 1) * 1024 + pd * 16 + pcol);
#else
      *(float4*)(sB + (cur ^ 1) * 1024 + pd * 16 + pcol) =
          *(const float4*)(gpanel + (c + 1) * 16);
#endif
    }

    const float* __restrict__ sBc = sB + cur * 1024;
    v8f acc = {};
#pragma unroll
    for (int kk = 0; kk < 16; ++kk) {
      const int d0 = 4 * kk + 2 * half;
      v2f A;  A.x = a0[kk];               A.y = a1[kk];
      v2f Bv; Bv.x = sBc[d0 * 16 + lcol]; Bv.y = sBc[(d0 + 1) * 16 + lcol];
      acc = __builtin_amdgcn_wmma_f32_16x16x4_f32(
          /*neg_a=*/false, A, /*neg_b=*/false, Bv,
          /*c_mod=*/(short)0, acc, /*reuse_a=*/false, /*reuse_b=*/false);
    }

    // epilogue: dist = a2 - 2*dot + b2, transposed store [B,K,S], argmin
    const float b2v = b2ws[c0 + lcol];
    float* __restrict__ dcol =
        dist + (size_t)(c0 + lcol) * S_ + s0 + 8 * half;  // 8 contiguous s
    const int myCol = c0 + lcol;
#pragma unroll
    for (int r = 0; r < 8; ++r) {
      const float dv = fmaf(-2.0f, acc[r], a2row[r] + b2v);
      dcol[r] = dv;
      if (dv < bestV[r]) { bestV[r] = dv; bestI[r] = myCol; }
    }

    // next panel must be fully resident in LDS before anyone reads it
#if USE_ASYNC_LDS
    wait_async0();
#endif
    __syncthreads();
  }

  // ---- argmin reduction across the 16 lanes of each half (ties -> low idx)
  int* __restrict__ oh   = (int*)out + OFF_OH + (size_t)b * K_ * S_;
  int* __restrict__ eidx = (int*)out + OFF_IDX + b * S_;
#pragma unroll
  for (int r = 0; r < 8; ++r) {
    float bv = bestV[r];
    int   bi = bestI[r];
#pragma unroll
    for (int m = 8; m >= 1; m >>= 1) {
      const float ov = __shfl_xor(bv, m, 32);
      const int   oi = __shfl_xor(bi, m, 32);
      if (ov < bv || (ov == bv && oi < bi)) { bv = ov; bi = oi; }
    }
    if (lcol == r) {                 // one lane per half writes its row
      const int srow = s0 + r + 8 * half;  // s index of row M = r+8*half
      eidx[srow] = bi;
      oh[(size_t)bi * S_ + srow] = 1;
    }
  }
}

// ---------------------------------------------------------------------------
// Kernel 4: gather codewords -> quantized_st, write unquantized, loss
// ---------------------------------------------------------------------------
__global__ void __launch_bounds__(256)
vq_gather(const float* __restrict__ x, const float* __restrict__ emb,
          float* __restrict__ out) {
  const int n = blockIdx.x * blockDim.x + threadIdx.x;  // = b*S + s
  const int b = n / S_, s = n % S_;
  const int idx = ((const int*)out)[OFF_IDX + n];

  const float* __restrict__ xc = x + (size_t)b * D_ * S_ + s;
  float* __restrict__ qc  = out + OFF_Q   + (size_t)b * D_ * S_ + s;
  float* __restrict__ unq = out + OFF_UNQ + (size_t)n * D_;

  float sum = 0.0f;
#pragma unroll 8
  for (int d = 0; d < D_; ++d) {
    const float e  = emb[(size_t)d * K_ + idx];
    const float xv = xc[(size_t)d * S_];
    qc[(size_t)d * S_] = e;        // quantized_st == quantized numerically
    unq[d] = xv;
    const float diff = e - xv;
    sum = fmaf(diff, diff, sum);
  }
  // loss = q_latent + e_latent = 2 * mean((q - x)^2)
#pragma unroll
  for (int m = 16; m >= 1; m >>= 1) sum += __shfl_xor(sum, m, 32);
  if ((threadIdx.x & 31) == 0)
    atomicAdd(out + OFF_LOSS, sum * (2.0f / (float)(BS_ * D_ * S_)));
}

// ---------------------------------------------------------------------------
extern "C" void kernel_launch(void* const* d_in, const int* in_sizes, int n_in,
                              void* d_out, int out_size, void* d_ws,
                              size_t ws_size, hipStream_t stream) {
  const float* x   = (const float*)d_in[0];   // [32,64,2048] f32
  const float* emb = (const float*)d_in[1];   // [64,1024]    f32
  float* out  = (float*)d_out;
  float* b2ws = (float*)d_ws;                 // K_ floats of scratch

  vq_prep<<<(K_ + 255) / 256, 256, 0, stream>>>(emb, out, b2ws);
  vq_zero_onehot<<<2048, 256, 0, stream>>>((int*)out + OFF_OH);
  vq_dist<<<(N_ / 16) / 8, 256, 0, stream>>>(x, emb, b2ws, out);
  vq_gather<<<N_ / 256, 256, 0, stream>>>(x, emb, out);
}